// RotaryEmbedding2D_65962107731996
// MI455X (gfx1250) — compile-verified
//
#include <hip/hip_runtime.h>
#include <cstdint>

// ----------------------------------------------------------------------------
// RotaryEmbedding2D: build (B,S,D,D) block-diagonal rotation matrices.
//   B=16, S=1024, D=64  -> 16384 positions, each a 64x64 fp32 matrix (16 KiB).
// Pure streaming-store kernel: 256 MiB out / 23.3 TB/s ~= 11.5 us floor.
// CDNA5 paths used: global_load_async_to_lds_b64 + s_wait_asynccnt (ASYNCcnt),
// non-temporal global_store_b128 for the 256 MiB stream.
// ----------------------------------------------------------------------------

typedef float v4f __attribute__((ext_vector_type(4)));

#define DD 64   // matrix dim; 32 rotation pairs; q = DD/4 = 16 freqs per axis

__global__ __launch_bounds__(256) void rope2d_build_R(
    const float* __restrict__ spa,   // (B*S, 2) fp32
    float* __restrict__ out)         // (B*S, 64, 64) fp32
{
    __shared__ float shXY[2];
    __shared__ float shSin[32];
    __shared__ float shCos[32];

    const int pos = blockIdx.x;      // 0 .. B*S-1, one 64x64 matrix per block

    // ---- stage spa_feat[pos][0:2] (8 bytes) into LDS via async DMA path ----
    if (threadIdx.x == 0) {
        unsigned long long gaddr =
            (unsigned long long)(uintptr_t)(spa + (size_t)pos * 2);
        unsigned int lds_off =
            (unsigned int)(uintptr_t)&shXY[0];   // low 32 bits = LDS offset
        asm volatile("global_load_async_to_lds_b64 %0, %1, off"
                     :
                     : "v"(lds_off), "v"(gaddr)
                     : "memory");
    }
    asm volatile("s_wait_asynccnt 0" ::: "memory");

    // ---- 32 lanes compute the 32 (sin,cos) pairs once, into LDS ----------
    if (threadIdx.x < 32) {
        const int h = threadIdx.x;               // rotation-pair index 0..31
        volatile float* vXY = shXY;
        const float coord = vXY[h >> 4];         // h<16 -> x, h>=16 -> y
        const float j = (float)(h & 15);
        const float invf = powf(10000.0f, -j * (1.0f / 16.0f));
        const float f = coord * invf;            // SCALE == 1
        shSin[h] = sinf(f);
        shCos[h] = cosf(f);
    }
    __syncthreads();

    // ---- stream the 64x64 matrix: 1024 float4 segments, NT b128 stores ----
    v4f* out4 = (v4f*)out + (size_t)pos * (DD * DD / 4);

#pragma unroll
    for (int it = 0; it < 4; ++it) {
        const int f4 = it * 256 + (int)threadIdx.x;  // 0..1023
        const int r  = f4 >> 4;                      // row 0..63
        const int c4 = f4 & 15;                      // float4 column 0..15
        const int h  = r >> 1;                       // pair index 0..31
        v4f v = {0.0f, 0.0f, 0.0f, 0.0f};
        if (c4 == (h >> 1)) {                        // the one nonzero segment
            const float s = shSin[h];
            const float c = shCos[h];
            float e0, e1;
            if (r & 1) { e0 = s; e1 = c;  }          // odd row:  [ sin  cos]
            else       { e0 = c; e1 = -s; }          // even row: [ cos -sin]
            if (h & 1) { v.z = e0; v.w = e1; }       // pair at cols 4k+2,4k+3
            else       { v.x = e0; v.y = e1; }       // pair at cols 4k,4k+1
        }
        __builtin_nontemporal_store(v, out4 + f4);
    }
}

extern "C" void kernel_launch(void* const* d_in, const int* in_sizes, int n_in,
                              void* d_out, int out_size, void* d_ws, size_t ws_size,
                              hipStream_t stream)
{
    const float* spa = (const float*)d_in[0];
    float* out = (float*)d_out;
    const int npos = in_sizes[0] / 2;      // B*S = 16384
    rope2d_build_R<<<npos, 256, 0, stream>>>(spa, out);
}